// InternalAWQuantizer_54872502173810
// MI455X (gfx1250) — compile-verified
//
#include <hip/hip_runtime.h>

typedef float v2f __attribute__((ext_vector_type(2)));
typedef float v8f __attribute__((ext_vector_type(8)));

#define OC_TOT   4096
#define IC_TOT   4096
#define NG       32        // IC / 128
#define GROUPSZ  128
#define TSTRIDE  4         // 512 tokens / 128 sampled
#define TS       128
#define NSTEPS   10
#define NGRID_F  20.0f

#define XPITCH   132       // padded LDS pitch (floats) -> conflict-free WMMA frag loads
#define WPITCH   132

// LDS budget (floats): Xs 128*132=16896, wT 16*132=2112, Ds 16*132=2112,
// redMax 256, redMin 256, scalars 6*16=96  => 21728 floats = 86912 bytes
#define LDS_FLOATS (128*XPITCH + 16*WPITCH + 16*WPITCH + 256 + 256 + 96)

__global__ __launch_bounds__(256)
void awq_autoclip_kernel(const float* __restrict__ w,
                         const float* __restrict__ xin,
                         float* __restrict__ out)
{
    extern __shared__ float lds[];
    float* Xs     = lds;                    // [128][XPITCH] token-major X_g
    float* wT     = Xs + 128 * XPITCH;      // [16][WPITCH]  w tile
    float* Ds     = wT + 16 * WPITCH;       // [16][WPITCH]  delta = q(w)-w
    float* redMax = Ds + 16 * WPITCH;       // [16 kt][16 oc]
    float* redMin = redMax + 256;
    float* smx    = redMin + 256;           // per-oc scalars (16 each)
    float* smn    = smx + 16;
    float* smaxv  = smn + 16;
    float* sscale = smaxv + 16;
    float* szero  = sscale + 16;
    float* serr   = szero + 16;

    const int tid = threadIdx.x;
    const int oc0 = blockIdx.x * 16;
    const int g   = blockIdx.y;

    // ---- stage W tile (16 oc x 128 k) into LDS ----
    {
        int fi0 = tid * 2;                       // 512 float4s total
        #pragma unroll
        for (int i = 0; i < 2; ++i) {
            int fi  = fi0 + i;
            int row = fi >> 5;                   // 0..15
            int k4  = (fi & 31) * 4;             // 0..124
            float4 v = *reinterpret_cast<const float4*>(
                w + (size_t)(oc0 + row) * IC_TOT + (size_t)g * GROUPSZ + k4);
            *reinterpret_cast<float4*>(&wT[row * WPITCH + k4]) = v;
        }
    }
    // ---- stage X_g (128 sampled tokens x 128 k) into LDS ----
    {
        int t    = tid >> 1;                     // 0..127
        int half = tid & 1;
        const float* src = xin + (size_t)(t * TSTRIDE) * IC_TOT
                               + (size_t)g * GROUPSZ + half * 64;
        float* dst = &Xs[t * XPITCH + half * 64];
        #pragma unroll
        for (int j = 0; j < 16; ++j)
            *reinterpret_cast<float4*>(dst + j * 4) =
                *reinterpret_cast<const float4*>(src + j * 4);
    }
    __syncthreads();

    // ---- per-oc group min/max of w ----
    const int oc = tid & 15;
    const int kt = tid >> 4;                     // 0..15, each owns 8 k's
    {
        float mx = -3.4e38f, mn = 3.4e38f;
        #pragma unroll
        for (int j = 0; j < 8; ++j) {
            float v = wT[oc * WPITCH + kt * 8 + j];
            mx = fmaxf(mx, v);
            mn = fminf(mn, v);
        }
        redMax[kt * 16 + oc] = mx;
        redMin[kt * 16 + oc] = mn;
    }
    __syncthreads();
    if (tid < 16) {
        float mx = -3.4e38f, mn = 3.4e38f;
        #pragma unroll
        for (int j = 0; j < 16; ++j) {
            mx = fmaxf(mx, redMax[j * 16 + tid]);
            mn = fminf(mn, redMin[j * 16 + tid]);
        }
        smx[tid] = mx;
        smn[tid] = mn;
    }
    __syncthreads();

    float minErr  = 1e9f;
    float bestMax = 0.0f;
    float orgMax  = 0.0f;
    if (tid < 16) {
        orgMax  = fmaxf(smx[tid], -smn[tid]);    // max |w| over group
        bestMax = orgMax;
    }

    const int lane = tid & 31;
    const int wave = tid >> 5;                   // 0..7 -> token tile
    const int mRow = lane & 15;                  // M (A) / N (B) index
    const int kHi  = (lane >> 4) * 2;            // K sub-offset for half-wave

    for (int s = 0; s < NSTEPS; ++s) {
        if (tid < 16) {
            float maxv  = orgMax * (1.0f - (float)s / NGRID_F);
            float cmx   = fminf(fmaxf(smx[tid], -maxv), maxv);
            float cmn   = fminf(fmaxf(smn[tid], -maxv), maxv);
            float scale = fmaxf(cmx - cmn, 1e-5f) * (1.0f / 15.0f);
            float zero  = fminf(fmaxf(rintf(-cmn / scale), 0.0f), 15.0f);
            smaxv[tid]  = maxv;
            sscale[tid] = scale;
            szero[tid]  = zero;
            serr[tid]   = 0.0f;
        }
        __syncthreads();

        // ---- quantize: Ds = q(clip(w)) - w ----
        {
            float maxv = smaxv[oc];
            float sc   = sscale[oc];
            float isc  = 1.0f / sc;
            float z    = szero[oc];
            #pragma unroll
            for (int h = 0; h < 2; ++h) {
                float4 wv4 = *reinterpret_cast<const float4*>(
                    &wT[oc * WPITCH + kt * 8 + h * 4]);
                float wa[4] = {wv4.x, wv4.y, wv4.z, wv4.w};
                float da[4];
                #pragma unroll
                for (int j = 0; j < 4; ++j) {
                    float wc = fminf(fmaxf(wa[j], -maxv), maxv);
                    float q  = fminf(fmaxf(rintf(wc * isc) + z, 0.0f), 15.0f);
                    da[j]    = (q - z) * sc - wa[j];
                }
                float4 d4 = make_float4(da[0], da[1], da[2], da[3]);
                *reinterpret_cast<float4*>(&Ds[oc * WPITCH + kt * 8 + h * 4]) = d4;
            }
        }
        __syncthreads();

        // ---- E(16 oc x 16 t) = D(16xK) * Xg^T(Kx16), K=128, via f32 WMMA ----
        v8f c = {};
        const int t0 = wave * 16;
        #pragma unroll 4
        for (int k0 = 0; k0 < GROUPSZ; k0 += 4) {
            v2f a = *reinterpret_cast<const v2f*>(&Ds[mRow * WPITCH + k0 + kHi]);
            v2f b = *reinterpret_cast<const v2f*>(&Xs[(t0 + mRow) * XPITCH + k0 + kHi]);
            c = __builtin_amdgcn_wmma_f32_16x16x4_f32(
                    false, a, false, b, (short)0, c, false, false);
        }

        // ---- sum of squares over token dim: per-half 16-lane xor reduction ----
        float v[8];
        #pragma unroll
        for (int r = 0; r < 8; ++r) v[r] = c[r] * c[r];
        #pragma unroll
        for (int r = 0; r < 8; ++r) {
            #pragma unroll
            for (int off = 1; off < 16; off <<= 1)
                v[r] += __shfl_xor(v[r], off, 32);
        }
        if ((lane & 15) == 0) {
            int ocBase = (lane >> 4) * 8;        // C layout: M = r + 8*half
            #pragma unroll
            for (int r = 0; r < 8; ++r)
                atomicAdd(&serr[ocBase + r], v[r]);
        }
        __syncthreads();

        if (tid < 16) {
            float err = serr[tid] * (1.0f / (float)TS);  // mean over tokens
            if (err < minErr) { minErr = err; bestMax = smaxv[tid]; }
        }
        __syncthreads();
    }

    if (tid < 16)
        out[(size_t)(oc0 + tid) * NG + g] = bestMax;
}

extern "C" void kernel_launch(void* const* d_in, const int* in_sizes, int n_in,
                              void* d_out, int out_size, void* d_ws, size_t ws_size,
                              hipStream_t stream) {
    (void)in_sizes; (void)n_in; (void)out_size; (void)d_ws; (void)ws_size;
    const float* w   = (const float*)d_in[0];
    const float* x   = (const float*)d_in[1];
    float*       out = (float*)d_out;

    dim3 grid(OC_TOT / 16, NG);
    dim3 block(256);
    size_t shmem = (size_t)LDS_FLOATS * sizeof(float);
    awq_autoclip_kernel<<<grid, block, shmem, stream>>>(w, x, out);
}